// MBPertTS_65000035058074
// MI455X (gfx1250) — compile-verified
//
#include <hip/hip_runtime.h>
#include <hip/hip_bf16.h>
#include <math.h>

// ---------------------------------------------------------------------------
// MBPertTS RK45 integrator for MI455X (gfx1250)
//   y' = y * (r + A@y + eps @ P[floor(T*t/30)])
// Dormand-Prince 4(5), rtol=1e-3 atol=1e-6, up to 512 adaptive steps.
//
// Matvec A@y uses V_WMMA_F32_16X16X4_F32:
//   A-operand  = y broadcast across all 16 rows (per-lane 2 consecutive y vals)
//   B-operand  = pre-swizzled tile of A^T (so N indexes matrix rows)
//   => every row of the 16x16 accumulator holds the same 16 dot products,
//      making extraction independent of the M-layout.
// 4 independent accumulators break the serial WMMA RAW chain (latency hiding);
// their elementwise sum is a fixed deterministic reassociation of the fp32
// accumulation. A (64 MB fp32) is L2-resident (192 MB L2).
// ---------------------------------------------------------------------------

typedef __attribute__((ext_vector_type(2))) float v2f;
typedef __attribute__((ext_vector_type(8))) float v8f;

#define NBLOCKS   32
#define NTHREADS  256
#define MAXSTEPS  512
#define RTOL_F    1e-3f
#define ATOL_F    1e-6f

__device__ __forceinline__ v2f mkv2(float a, float b) { v2f v; v.x = a; v.y = b; return v; }

// ---------------- grid-wide barrier (sense-reversal via generation) --------
__device__ __forceinline__ void grid_barrier(volatile int* cnt, volatile int* gen) {
  __threadfence();          // make this block's writes device-visible
  __syncthreads();
  if (threadIdx.x == 0) {
    int g = *gen;
    if (atomicAdd((int*)cnt, 1) == NBLOCKS - 1) {
      *cnt = 0;             // all arrived; safe to reset
      __threadfence();
      atomicAdd((int*)gen, 1);
    } else {
      while (*gen == g) { __builtin_amdgcn_s_sleep(1); }
    }
  }
  __syncthreads();
  __threadfence();          // acquire: see other blocks' writes
}

// ---------------- pack A into WMMA B-operand layout ------------------------
// packed float4 index g = ((cb * ktiles) + i) * 32 + lane
//   lane<16 : koff=0, lane>=16 : koff=2 ; col = cb*16 + (lane&15)
//   .x=A[col][kb+koff] .y=A[col][kb+koff+1] .z=A[col][kb+4+koff] .w=A[col][kb+4+koff+1]
__global__ void prep_pack(const float* __restrict__ A, float* __restrict__ packed,
                          int n, int* __restrict__ bar) {
  size_t g = (size_t)blockIdx.x * blockDim.x + threadIdx.x;
  if (g == 0) { bar[0] = 0; bar[1] = 0; }   // reset grid-barrier state each call
  size_t total = ((size_t)n * n) >> 2;
  if (g >= total) return;
  int lane   = (int)(g & 31);
  int ktiles = n >> 3;
  int i      = (int)((g >> 5) % (size_t)ktiles);
  int cb     = (int)(g / ((size_t)ktiles << 5));
  int koff   = (lane >= 16) ? 2 : 0;
  int col    = cb * 16 + (lane & 15);
  int kb     = i * 8;
  const float* arow = A + (size_t)col * n;
  float4 v;
  v.x = arow[kb + koff];
  v.y = arow[kb + koff + 1];
  v.z = arow[kb + 4 + koff];
  v.w = arow[kb + 4 + koff + 1];
  ((float4*)packed)[g] = v;
}

// ---------------- persistent RK45 kernel -----------------------------------
__global__ void __launch_bounds__(NTHREADS)
mbpert_rk45(const float* __restrict__ x0, const int* __restrict__ tptr,
            const float* __restrict__ growth, const float* __restrict__ eps,
            const float* __restrict__ Pmat,
            const float* __restrict__ packed,
            float* __restrict__ ysg, float* __restrict__ Ay,
            float* __restrict__ blockSums, int* __restrict__ bar,
            float* __restrict__ out, int n, int perts, int T) {
  __shared__ float s_ys[4096];
  __shared__ float s_red[NTHREADS];

  const int tid  = threadIdx.x;
  const int blk  = blockIdx.x;
  const int lane = tid & 31;
  const int wave = tid >> 5;
  const int gw   = blk * (NTHREADS / 32) + wave;       // global wave id 0..255
  const int totWaves = NBLOCKS * (NTHREADS / 32);
  const int cbCount  = n >> 4;                         // 16-column blocks
  const int ktiles   = n >> 3;                         // 8 K-values per tile pair
  const int rpb      = n / NBLOCKS;                    // rows per block (128)
  const bool owner   = tid < rpb;
  const int row      = blk * rpb + tid;

  volatile int* cnt = bar;
  volatile int* gen = bar + 1;

  const float t_end = (float)(*tptr);
  float tcur = 0.0f;
  float h    = t_end * 0.01f;

  // ---- per-row register state (owner threads) ----
  float yR = 0.0f, rR = 0.0f, ynewR = 0.0f;
  float epsR[8];
  float kR[7];
  #pragma unroll
  for (int j = 0; j < 8; ++j) epsR[j] = 0.0f;
  #pragma unroll
  for (int j = 0; j < 7; ++j) kR[j] = 0.0f;
  if (owner) {
    yR = x0[row];
    rR = growth[row];
    for (int j = 0; j < perts && j < 8; ++j) epsR[j] = eps[row * perts + j];
  }

  for (int step = 0; step < MAXSTEPS; ++step) {
    h = fminf(h, t_end - tcur);
    if (!(tcur < t_end)) break;     // all blocks/threads agree (deterministic)

    float ysR = 0.0f;
    for (int s = 0; s < 7; ++s) {
      // -- phase 1: build stage input vector ys ---------------------------
      if (owner) {
        float acc = 0.0f;
        switch (s) {
          case 0: break;
          case 1: acc = (float)(1.0/5.0)*kR[0]; break;
          case 2: acc = (float)(3.0/40.0)*kR[0] + (float)(9.0/40.0)*kR[1]; break;
          case 3: acc = (float)(44.0/45.0)*kR[0] - (float)(56.0/15.0)*kR[1]
                      + (float)(32.0/9.0)*kR[2]; break;
          case 4: acc = (float)(19372.0/6561.0)*kR[0] - (float)(25360.0/2187.0)*kR[1]
                      + (float)(64448.0/6561.0)*kR[2] - (float)(212.0/729.0)*kR[3]; break;
          case 5: acc = (float)(9017.0/3168.0)*kR[0] - (float)(355.0/33.0)*kR[1]
                      + (float)(46732.0/5247.0)*kR[2] + (float)(49.0/176.0)*kR[3]
                      - (float)(5103.0/18656.0)*kR[4]; break;
          default: acc = (float)(35.0/384.0)*kR[0] + (float)(500.0/1113.0)*kR[2]
                       + (float)(125.0/192.0)*kR[3] - (float)(2187.0/6784.0)*kR[4]
                       + (float)(11.0/84.0)*kR[5]; break;
        }
        ysR = (s == 0) ? yR : (yR + h * acc);
        if (s == 6) ynewR = ysR;     // 5th-order solution y5 (FSAL row)
        ysg[row] = ysR;
      }
      grid_barrier(cnt, gen);

      // -- phase 2: stage ys into LDS, then WMMA matvec -------------------
      for (int it = tid; it < n; it += NTHREADS) s_ys[it] = ysg[it];
      __syncthreads();

      for (int cb = gw; cb < cbCount; cb += totWaves) {
        v8f c0 = {0.f,0.f,0.f,0.f,0.f,0.f,0.f,0.f};
        v8f c1 = c0, c2 = c0, c3 = c0;
        const float4* pb = (const float4*)packed + (size_t)cb * ktiles * 32 + lane;
        const int koff = (lane >= 16) ? 2 : 0;
        // 4 independent WMMA accumulation chains; unroll-by-4 over 8-K tiles
        for (int i = 0; i < ktiles; i += 4) {
          if ((i & 15) == 0)  // speculative L2 prefetch ahead
            __builtin_prefetch(&pb[(size_t)(i + 16) * 32], 0, 1);
          float4 t0 = pb[(size_t)(i + 0) * 32];   // global_load_b128 (clause)
          float4 t1 = pb[(size_t)(i + 1) * 32];
          float4 t2 = pb[(size_t)(i + 2) * 32];
          float4 t3 = pb[(size_t)(i + 3) * 32];
          const float* yp = &s_ys[i * 8 + koff];
          float2 y00 = *(const float2*)(yp +  0);  // ds_load_2addr_b64 pairs
          float2 y01 = *(const float2*)(yp +  4);
          float2 y10 = *(const float2*)(yp +  8);
          float2 y11 = *(const float2*)(yp + 12);
          float2 y20 = *(const float2*)(yp + 16);
          float2 y21 = *(const float2*)(yp + 20);
          float2 y30 = *(const float2*)(yp + 24);
          float2 y31 = *(const float2*)(yp + 28);
          c0 = __builtin_amdgcn_wmma_f32_16x16x4_f32(false, mkv2(y00.x, y00.y),
                 false, mkv2(t0.x, t0.y), (short)0, c0, false, false);
          c1 = __builtin_amdgcn_wmma_f32_16x16x4_f32(false, mkv2(y10.x, y10.y),
                 false, mkv2(t1.x, t1.y), (short)0, c1, false, false);
          c2 = __builtin_amdgcn_wmma_f32_16x16x4_f32(false, mkv2(y20.x, y20.y),
                 false, mkv2(t2.x, t2.y), (short)0, c2, false, false);
          c3 = __builtin_amdgcn_wmma_f32_16x16x4_f32(false, mkv2(y30.x, y30.y),
                 false, mkv2(t3.x, t3.y), (short)0, c3, false, false);
          c0 = __builtin_amdgcn_wmma_f32_16x16x4_f32(false, mkv2(y01.x, y01.y),
                 false, mkv2(t0.z, t0.w), (short)0, c0, false, false);
          c1 = __builtin_amdgcn_wmma_f32_16x16x4_f32(false, mkv2(y11.x, y11.y),
                 false, mkv2(t1.z, t1.w), (short)0, c1, false, false);
          c2 = __builtin_amdgcn_wmma_f32_16x16x4_f32(false, mkv2(y21.x, y21.y),
                 false, mkv2(t2.z, t2.w), (short)0, c2, false, false);
          c3 = __builtin_amdgcn_wmma_f32_16x16x4_f32(false, mkv2(y31.x, y31.y),
                 false, mkv2(t3.z, t3.w), (short)0, c3, false, false);
        }
        // fixed deterministic combine of the 4 partial accumulators
        v8f c = (c0 + c1) + (c2 + c3);
        // all M rows of D are identical; N = lane selects the output column
        if (lane < 16) Ay[cb * 16 + lane] = c[0];
      }
      grid_barrier(cnt, gen);

      // -- phase 3: k_s = ys * (r + A@ys + eps@P[d]) ----------------------
      if (owner) {
        float ts;
        switch (s) {                   // match reference arithmetic exactly
          case 0: ts = tcur; break;
          case 1: ts = tcur + h / 5.0f; break;
          case 2: ts = tcur + 3.0f * h / 10.0f; break;
          case 3: ts = tcur + 4.0f * h / 5.0f; break;
          case 4: ts = tcur + 8.0f * h / 9.0f; break;
          default: ts = tcur + h; break;
        }
        int d = (int)(((float)T * ts) / 30.0f);
        d = d < 0 ? 0 : (d > T ? T : d);
        float pv = 0.0f;
        for (int j = 0; j < perts && j < 8; ++j) pv += epsR[j] * Pmat[d * perts + j];
        kR[s] = ysR * (rR + Ay[row] + pv);
      }
    }

    // -- error estimate + deterministic global reduction ------------------
    float sq = 0.0f;
    if (owner) {
      float err = h * ((float)(71.0/57600.0)*kR[0] - (float)(71.0/16695.0)*kR[2]
                     + (float)(71.0/1920.0)*kR[3] - (float)(17253.0/339200.0)*kR[4]
                     + (float)(22.0/525.0)*kR[5] - (float)(1.0/40.0)*kR[6]);
      float sc = ATOL_F + RTOL_F * fmaxf(fabsf(yR), fabsf(ynewR));
      float q = err / sc;
      sq = q * q;
    }
    s_red[tid] = sq;
    __syncthreads();
    #pragma unroll
    for (int off = NTHREADS / 2; off > 0; off >>= 1) {
      if (tid < off) s_red[tid] += s_red[tid + off];
      __syncthreads();
    }
    if (tid == 0) blockSums[blk] = s_red[0];
    grid_barrier(cnt, gen);

    float tot = 0.0f;
    {
      volatile const float* bs = blockSums;   // fixed order -> identical everywhere
      for (int b = 0; b < NBLOCKS; ++b) tot += bs[b];
    }
    float err_norm = sqrtf(tot / (float)n);
    bool accept = (err_norm <= 1.0f);
    if (accept) {
      tcur += h;
      if (owner) yR = ynewR;
    }
    float factor = 0.9f * powf(err_norm + 1e-10f, -0.2f);
    factor = fminf(fmaxf(factor, 0.2f), 10.0f);
    h = h * factor;                   // active is always true inside the loop
  }

  if (owner) out[row] = yR;
}

// ---------------------------------------------------------------------------
extern "C" void kernel_launch(void* const* d_in, const int* in_sizes, int n_in,
                              void* d_out, int out_size, void* d_ws, size_t ws_size,
                              hipStream_t stream) {
  const float* x    = (const float*)d_in[0];
  const int*   tptr = (const int*)  d_in[1];
  const float* r    = (const float*)d_in[2];
  const float* A    = (const float*)d_in[3];
  const float* eps  = (const float*)d_in[4];
  const float* P    = (const float*)d_in[5];

  const int n     = in_sizes[0];                 // 4096
  const int perts = in_sizes[4] / n;             // 8
  const int T     = in_sizes[5] / perts - 1;     // 30

  // workspace layout
  float* packed    = (float*)d_ws;               // n*n floats (64 MB)
  float* ysg       = packed + (size_t)n * n;     // n
  float* Ay        = ysg + n;                    // n
  float* blockSums = Ay + n;                     // NBLOCKS
  int*   bar       = (int*)(blockSums + NBLOCKS);// 2 ints

  // 1) swizzle A into WMMA-ready layout + reset barrier state (every call)
  {
    size_t total = ((size_t)n * n) >> 2;
    int blocks = (int)((total + 255) / 256);
    prep_pack<<<blocks, 256, 0, stream>>>(A, packed, n, bar);
  }

  // 2) persistent adaptive-RK45 integrator
  mbpert_rk45<<<NBLOCKS, NTHREADS, 0, stream>>>(
      x, tptr, r, eps, P, packed, ysg, Ay, blockSums, bar,
      (float*)d_out, n, perts, T);
}